// image_prj_46651934769375
// MI455X (gfx1250) — compile-verified
//
#include <hip/hip_runtime.h>
#include <math.h>

// ---- problem constants (match reference exactly) ----
#define IMG_M   1024         // input side
#define HH      512          // DWT output side (M/2)
#define PADT    213          // ceil(sqrt(2)*512 - 512)
#define PB      106          // pad before
#define PP      725          // padded side (HH + PADT)
#define NANG    180
#define CENTERF 362.0f       // float(PP // 2)
#define THRESH  0.075f
#define NTILES  ((PP + 15) / 16)   // 46 column tiles of 16

typedef float v2f __attribute__((ext_vector_type(2)));
typedef float v8f __attribute__((ext_vector_type(8)));

// ---------------------------------------------------------------------------
// Kernel 1: 2x2 stride-2 high-pass conv + hard threshold + zero pad to PP x PP
// ---------------------------------------------------------------------------
__global__ void dwt_pad_kernel(const float* __restrict__ image,
                               float* __restrict__ img /* PP*PP */) {
    int idx = blockIdx.x * blockDim.x + threadIdx.x;
    if (idx >= PP * PP) return;
    int i = idx / PP;
    int j = idx - i * PP;
    float v = 0.0f;
    int hi = i - PB;
    int hj = j - PB;
    if (hi >= 0 && hi < HH && hj >= 0 && hj < HH) {
        const float* r0 = image + (2 * hi) * IMG_M + 2 * hj;
        const float* r1 = r0 + IMG_M;
        float a = r0[0], b = r0[1], c = r1[0], d = r1[1];
        v = fmaf(-0.1384f, a,
            fmaf( 0.7243f, b,
            fmaf(-0.6038f, c, 0.1601f * d)));
        v = (fabsf(v) >= THRESH) ? v : 0.0f;
    }
    img[idx] = v;
}

// ---------------------------------------------------------------------------
// Bilinear sample with WRAP fill mode (periodic integer coords)
// ---------------------------------------------------------------------------
__device__ __forceinline__ float bilin_wrap(const float* __restrict__ img,
                                            float bx, float by,
                                            float s, float c, int y) {
    float yf  = (float)y;
    float xin = fmaf(s, yf, bx);
    float yin = fmaf(c, yf, by);
    float xfl = floorf(xin);
    float yfl = floorf(yin);
    float fx  = xin - xfl;
    float fy  = yin - yfl;
    int ix = (int)xfl;
    int iy = (int)yfl;
    int x0 = ix % PP; if (x0 < 0) x0 += PP;
    int y0 = iy % PP; if (y0 < 0) y0 += PP;
    int x1 = x0 + 1;  if (x1 == PP) x1 = 0;
    int y1 = y0 + 1;  if (y1 == PP) y1 = 0;
    const float* r0 = img + y0 * PP;
    const float* r1 = img + y1 * PP;
    float v00 = r0[x0], v01 = r0[x1];
    float v10 = r1[x0], v11 = r1[x1];
    float top = fmaf(fx, v01 - v00, v00);
    float bot = fmaf(fx, v11 - v10, v10);
    return fmaf(fy, bot - top, top);
}

// ---------------------------------------------------------------------------
// Kernel 2: Radon line integrals. One wave32 per (angle, 16-column tile).
// Lane layout: colIdx = lane & 15, ysel = lane >> 4. Per y-step of 4, each
// lane accumulates 2 samples -> 4 f32 partials per column across the wave,
// arranged exactly as a 4x16 B matrix for V_WMMA_F32_16X16X4_F32. One WMMA
// with all-ones A collapses K -> column sums (D row 0, lanes 0..15).
// ---------------------------------------------------------------------------
__global__ void radon_kernel(const float* __restrict__ img,
                             const float* __restrict__ theta,
                             float* __restrict__ out /* PP x NANG */) {
    int tile = blockIdx.x;          // 0..NTILES-1
    int ang  = blockIdx.y;          // 0..NANG-1
    int lane = threadIdx.x;         // 0..31
    int colIdx = lane & 15;
    int ysel   = lane >> 4;         // 0 or 1
    int col    = tile * 16 + colIdx;

    float a = theta[ang];           // raw radians, per reference
    float c = cosf(a);
    float s = sinf(a);
    float colf = (float)col;
    // xin = c*x + s*y - center*(c+s-1) ; yin = -s*x + c*y - center*(c-s-1)
    float bx = fmaf(c,  colf, -CENTERF * (c + s - 1.0f));
    float by = fmaf(-s, colf, -CENTERF * (c - s - 1.0f));

    // accA -> K = 2*ysel, accB -> K = 2*ysel + 1 of the 4x16 B matrix
    float accA = 0.0f;
    float accB = 0.0f;
    for (int yb = 0; yb < PP; yb += 4) {
        int y1 = yb + 2 * ysel;
        int y2 = y1 + 1;
        if (y1 < PP) accA += bilin_wrap(img, bx, by, s, c, y1);
        if (y2 < PP) accB += bilin_wrap(img, bx, by, s, c, y2);
    }

    // D(16x16) = ones(16x4) x B(4x16) + 0  => every row of D = column sums
    v2f A; A.x = 1.0f; A.y = 1.0f;     // all-ones A: layout-independent
    v2f B; B.x = accA; B.y = accB;     // VGPR0: K=0(l0-15)/K=2(l16-31); VGPR1: K=1/K=3
    v8f C = {};
    v8f D = __builtin_amdgcn_wmma_f32_16x16x4_f32(
        /*neg_a=*/false, A, /*neg_b=*/false, B,
        /*c_mod=*/(short)0, C, /*reuse_a=*/false, /*reuse_b=*/false);

    float colsum = D[0];               // lanes 0..15: D[M=0][N=lane] = sum_K B[K][lane]
    if (lane < 16 && col < PP) {
        out[col * NANG + ang] = colsum;   // A.T layout: (P, n_angles)
    }
}

// ---------------------------------------------------------------------------
// Kernel 3: global max over unnormalized sinogram (single block, LDS tree)
// ---------------------------------------------------------------------------
__global__ void reduce_max_kernel(const float* __restrict__ a, int n,
                                  float* __restrict__ maxv) {
    __shared__ float sm[1024];
    float m = -INFINITY;
    for (int i = threadIdx.x; i < n; i += blockDim.x) m = fmaxf(m, a[i]);
    sm[threadIdx.x] = m;
    __syncthreads();
    for (int off = 512; off > 0; off >>= 1) {
        if ((int)threadIdx.x < off)
            sm[threadIdx.x] = fmaxf(sm[threadIdx.x], sm[threadIdx.x + off]);
        __syncthreads();
    }
    if (threadIdx.x == 0) *maxv = sm[0];
}

// ---------------------------------------------------------------------------
// Kernel 4: normalize in place (d_out fully rewritten by kernel 2 each call,
// so in-place divide is replay-safe and deterministic)
// ---------------------------------------------------------------------------
__global__ void normalize_kernel(float* __restrict__ out, int n,
                                 const float* __restrict__ maxv) {
    int i = blockIdx.x * blockDim.x + threadIdx.x;
    if (i < n) out[i] = out[i] / (*maxv);
}

// ---------------------------------------------------------------------------
extern "C" void kernel_launch(void* const* d_in, const int* in_sizes, int n_in,
                              void* d_out, int out_size, void* d_ws, size_t ws_size,
                              hipStream_t stream) {
    const float* image = (const float*)d_in[0];   // (1,1024,1024) f32
    const float* theta = (const float*)d_in[1];   // (180,) f32
    float* out  = (float*)d_out;                  // (725,180) f32
    float* img  = (float*)d_ws;                   // PP*PP padded image
    float* maxv = img + (size_t)PP * PP;          // 1 float scalar

    // 1) DWT high-pass + threshold + pad
    {
        int n = PP * PP;
        dwt_pad_kernel<<<(n + 255) / 256, 256, 0, stream>>>(image, img);
    }
    // 2) Radon with WMMA f32 row-sum reduction
    {
        dim3 grid(NTILES, NANG);
        radon_kernel<<<grid, 32, 0, stream>>>(img, theta, out);
    }
    // 3) global max
    reduce_max_kernel<<<1, 1024, 0, stream>>>(out, PP * NANG, maxv);
    // 4) normalize
    {
        int n = PP * NANG;
        normalize_kernel<<<(n + 255) / 256, 256, 0, stream>>>(out, n, maxv);
    }
}